// HybridQuantumNet_83030307766338
// MI455X (gfx1250) — compile-verified
//
#include <hip/hip_runtime.h>
#include <math.h>

// MI455X / gfx1250. Quantum circuit collapses to: logit = w0^T H w1 with
// w = [cos x, sin x, 1]; H (3x3) precomputed from q_weights/fc on device.
// Main kernel: memory-bound stream (12 B/sample, ~25 MB total), bilinear
// form evaluated as diag(A*B) via V_WMMA_F32_16X16X4_F32, 16 samples/wave/step.

typedef __attribute__((ext_vector_type(2))) float v2f;
typedef __attribute__((ext_vector_type(8))) float v8f;

#define QDEPTH 5

// ---- one-thread setup: build U = prod_l CNOT*(RX(w_l0) (x) RX(w_l1)),
// S = Re(U^H (Z(x)I) U), contract half-angle quadratics to 3x3 H, fold fc. ----
__global__ void qnet_setup_kernel(const float* __restrict__ qw,
                                  const float* __restrict__ fcw,
                                  const float* __restrict__ fcb,
                                  float* __restrict__ Hout) {
  if (threadIdx.x != 0 || blockIdx.x != 0) return;
  float Ur[4][4] = {}, Ui[4][4] = {};
  Ur[0][0] = Ur[1][1] = Ur[2][2] = Ur[3][3] = 1.0f;
  for (int l = 0; l < QDEPTH; ++l) {
    float c0 = cosf(qw[2 * l] * 0.5f),     s0 = sinf(qw[2 * l] * 0.5f);
    float c1 = cosf(qw[2 * l + 1] * 0.5f), s1 = sinf(qw[2 * l + 1] * 0.5f);
    // RX on qubit0: partner row flips bit1 of flat index (i = 2*q0 + q1)
    for (int j = 0; j < 4; ++j) {
      float nr[4], ni[4];
      for (int i = 0; i < 4; ++i) {
        int o = i ^ 2;  // (-i*s)*(xr+i*xi) = s*xi - i*s*xr
        nr[i] = c0 * Ur[i][j] + s0 * Ui[o][j];
        ni[i] = c0 * Ui[i][j] - s0 * Ur[o][j];
      }
      for (int i = 0; i < 4; ++i) { Ur[i][j] = nr[i]; Ui[i][j] = ni[i]; }
    }
    // RX on qubit1: partner flips bit0
    for (int j = 0; j < 4; ++j) {
      float nr[4], ni[4];
      for (int i = 0; i < 4; ++i) {
        int o = i ^ 1;
        nr[i] = c1 * Ur[i][j] + s1 * Ui[o][j];
        ni[i] = c1 * Ui[i][j] - s1 * Ur[o][j];
      }
      for (int i = 0; i < 4; ++i) { Ur[i][j] = nr[i]; Ui[i][j] = ni[i]; }
    }
    // CNOT(0->1): swap rows 2 and 3
    for (int j = 0; j < 4; ++j) {
      float tr = Ur[2][j]; Ur[2][j] = Ur[3][j]; Ur[3][j] = tr;
      float ti = Ui[2][j]; Ui[2][j] = Ui[3][j]; Ui[3][j] = ti;
    }
  }
  // S = Re(U^H diag(1,1,-1,-1) U)  (symmetric)
  float S[4][4];
  for (int i = 0; i < 4; ++i)
    for (int j = 0; j < 4; ++j) {
      float acc = 0.0f;
      for (int k = 0; k < 4; ++k) {
        float z = (k < 2) ? 1.0f : -1.0f;
        acc += z * (Ur[k][i] * Ur[k][j] + Ui[k][i] * Ui[k][j]);
      }
      S[i][j] = acc;
    }
  // For each (a,b): 2x2 block over q1 -> g_ab dotted with [cos x1, sin x1, 1]
  float g[2][2][3];
  for (int a = 0; a < 2; ++a)
    for (int b = 0; b < 2; ++b) {
      float B00 = S[2 * a + 0][2 * b + 0], B01 = S[2 * a + 0][2 * b + 1];
      float B10 = S[2 * a + 1][2 * b + 0], B11 = S[2 * a + 1][2 * b + 1];
      g[a][b][0] = 0.5f * (B00 - B11);
      g[a][b][1] = 0.5f * (B01 + B10);
      g[a][b][2] = 0.5f * (B00 + B11);
    }
  float w = fcw[0], b = fcb[0];
  float hh[9];
  for (int j = 0; j < 3; ++j) {
    hh[0 * 3 + j] = w * 0.5f * (g[0][0][j] - g[1][1][j]);  // pairs with cos x0
    hh[1 * 3 + j] = w * 0.5f * (g[0][1][j] + g[1][0][j]);  // pairs with sin x0
    hh[2 * 3 + j] = w * 0.5f * (g[0][0][j] + g[1][1][j]);  // pairs with 1
  }
  hh[8] += b;  // fold fc bias into constant*constant term
  for (int i = 0; i < 9; ++i) Hout[i] = hh[i];
}

// ---- main streaming kernel: 16 samples per wave per iteration via WMMA ----
__global__ void __launch_bounds__(256)
qnet_wmma_kernel(const float2* __restrict__ x, const float* __restrict__ H,
                 float* __restrict__ out, int n, int nTiles) {
  const int lane   = threadIdx.x & 31;
  const int wave   = (int)((blockIdx.x * blockDim.x + threadIdx.x) >> 5);
  const int nWaves = (int)((gridDim.x * blockDim.x) >> 5);
  const bool low   = lane < 16;

  const float H00 = H[0], H01 = H[1], H02 = H[2];
  const float H10 = H[3], H11 = H[4], H12 = H[5];
  const float H20 = H[6], H21 = H[7], H22 = H[8];

  // D diag ownership: lanes 0-7 hold D[L][L] in vgpr L; lanes 24-31 hold
  // D[L-16][L-16] in vgpr L-24. Others hold no diagonal element.
  const int  k     = low ? lane : (lane - 24);
  const bool valid = (lane < 8) || (lane >= 24);

  for (int tile = wave; tile < nTiles; tile += nWaves) {
    int s  = tile * 16 + (lane & 15);     // both half-waves mirror samples
    int sc = (s < n) ? s : (n - 1);       // clamp (EXEC must stay full for WMMA)
    float2 xx = x[sc];
    float c0 = __cosf(xx.x), s0 = __sinf(xx.x);
    float c1 = __cosf(xx.y), s1 = __sinf(xx.y);

    // A (16x4): row m = [cos x0_m, sin x0_m, 1, 0]
    v2f A;
    A.x = low ? c0 : 1.0f;   // K=0 (low) / K=2 (high)
    A.y = low ? s0 : 0.0f;   // K=1 (low) / K=3 (high)
    // B (4x16): column n = [H0.w1, H1.w1, H2.w1, 0], w1 = [c1, s1, 1]
    float bl0 = fmaf(H00, c1, fmaf(H01, s1, H02));
    float bl1 = fmaf(H10, c1, fmaf(H11, s1, H12));
    float bh0 = fmaf(H20, c1, fmaf(H21, s1, H22));
    v2f B;
    B.x = low ? bl0 : bh0;   // row K=0 (low) / K=2 (high)
    B.y = low ? bl1 : 0.0f;  // row K=1 (low) / K=3 (high)

    v8f c = {};
    v8f d = __builtin_amdgcn_wmma_f32_16x16x4_f32(
        /*neg_a=*/false, A, /*neg_b=*/false, B,
        /*c_mod=*/(short)0, c, /*reuse_a=*/false, /*reuse_b=*/false);

    // pick my diagonal element D[m][m] = logit for sample m
    float t = d[0];
    t = (k == 1) ? d[1] : t;
    t = (k == 2) ? d[2] : t;
    t = (k == 3) ? d[3] : t;
    t = (k == 4) ? d[4] : t;
    t = (k == 5) ? d[5] : t;
    t = (k == 6) ? d[6] : t;
    t = (k == 7) ? d[7] : t;

    // sigmoid via hardware exp + hardware reciprocal (v_exp_f32 + v_rcp_f32)
    float r = __builtin_amdgcn_rcpf(1.0f + __expf(-t));
    if (valid && (s < n)) out[s] = r;
  }
}

extern "C" void kernel_launch(void* const* d_in, const int* in_sizes, int n_in,
                              void* d_out, int out_size, void* d_ws, size_t ws_size,
                              hipStream_t stream) {
  const float* x   = (const float*)d_in[0];  // [N,2]
  const float* qw  = (const float*)d_in[1];  // [5,2]
  const float* fcw = (const float*)d_in[2];  // [1,1]
  const float* fcb = (const float*)d_in[3];  // [1]
  float* out = (float*)d_out;
  float* H   = (float*)d_ws;                 // 9 floats scratch

  int n = in_sizes[0] / 2;
  qnet_setup_kernel<<<1, 32, 0, stream>>>(qw, fcw, fcb, H);

  int nTiles = (n + 15) / 16;
  const int TILES_PER_WAVE = 8;
  int waves  = (nTiles + TILES_PER_WAVE - 1) / TILES_PER_WAVE;
  int blocks = (waves + 7) / 8;  // 256 threads = 8 waves per block
  if (blocks < 1) blocks = 1;
  qnet_wmma_kernel<<<blocks, 256, 0, stream>>>((const float2*)x, H, out, n, nTiles);
}